// PoseLoss_80831284510874
// MI455X (gfx1250) — compile-verified
//
#include <hip/hip_runtime.h>
#include <math.h>

typedef __attribute__((ext_vector_type(2))) float v2f;
typedef __attribute__((ext_vector_type(8))) float v8f;

#define KPT 15
#define ZBN 64
#define HH 64
#define WW 64
#define CHN 79            // KPT + ZBN
#define BB 32
#define NCH 8             // chunks per image in k1
#define PIXC (HH*WW/NCH)  // 512 pixels per chunk (8 rows)
#define MINL (-150.0f)
#define DXY (300.0f/63.0f)

__constant__ int d_limb[KPT] = {0,0,1,1,1,3,4,5,6,2,2,9,10,11,12};
__constant__ int d_sym[KPT]  = {0,1,2,4,3,6,5,8,7,10,9,12,11,14,13};

// ---------------------------------------------------------------------------
// K1: streaming spatial-softmax moment GEMM via V_WMMA_F32_16X16X4_F32
//   out[k,c] = sum_n exp(f[n,k]) * locext[n,c],  locext = (1, x, y, x^2+y^2)
// One workgroup = one (image, chunk of 8 rows). Each wave owns one row (x is
// a wave constant), 16 WMMAs/wave over 64 pixels. Loop body is branchless:
// the moment-column selection uses per-lane coefficients hoisted out of the
// loop so the unrolled loads can all issue together.
// ---------------------------------------------------------------------------
__global__ __launch_bounds__(256) void k1_xy_partial(const float* __restrict__ feat,
                                                     float* __restrict__ part)
{
    const int wg   = blockIdx.x;            // 0 .. BB*NCH-1
    const int b    = wg / NCH;
    const int ch   = wg % NCH;
    const int lane = threadIdx.x & 31;
    const int wave = threadIdx.x >> 5;
    const int half = lane >> 4;             // 0: lanes 0-15, 1: lanes 16-31
    const int m    = lane & 15;             // A: channel row, B: moment column

    const int   row  = ch * 8 + wave;       // this wave's image row (h)
    const float xrow = MINL + row * DXY;    // x coordinate: wave constant

    // per-lane selector coefficients (loop-invariant, branchless body)
    const float c1    = (m == 0)  ? 1.0f : 0.0f;
    const float cx    = (m == 1)  ? 1.0f : 0.0f;
    const float cy    = (m == 2)  ? 1.0f : 0.0f;
    const float cr    = (m == 3)  ? 1.0f : 0.0f;
    const float emask = (m == 15) ? 0.0f : 1.0f;   // channel-15 pad row of A
    const float bconst = c1 + cx * xrow + cr * (xrow * xrow);

    // row base pointer for this lane's channel
    const float* fr = feat + ((size_t)(b * HH + row) * WW) * CHN + m;

    v8f acc = {};
    #pragma unroll 4
    for (int g = 0; g < 16; ++g) {               // 16 groups of 4 pixels
        const int w0 = g * 4 + 2 * half;         // WMMA K=0/2 pixel (col w)
        const float f0 = fr[w0 * CHN];           // K=0/2
        const float f1 = fr[w0 * CHN + CHN];     // K=1/3
        v2f A;
        A.x = emask * __expf(f0);
        A.y = emask * __expf(f1);

        const float y0 = MINL + (float)w0 * DXY;
        const float y1 = y0 + DXY;
        v2f Bm;
        Bm.x = bconst + cy * y0 + cr * (y0 * y0);
        Bm.y = bconst + cy * y1 + cr * (y1 * y1);

        acc = __builtin_amdgcn_wmma_f32_16x16x4_f32(false, A, false, Bm,
                                                    (short)0, acc, false, false);
    }

    // D layout: vgpr j, lanes 0-15 -> (k=j, c=lane), lanes 16-31 -> (k=j+8, c=lane-16)
    __shared__ float red[8][64];
    if (m < 4) {
        #pragma unroll
        for (int j = 0; j < 8; ++j) {
            const int k = j + 8 * half;
            red[wave][k * 4 + m] = acc[j];
        }
    }
    __syncthreads();
    if (threadIdx.x < 64) {
        float s = 0.0f;
        #pragma unroll
        for (int wv = 0; wv < 8; ++wv) s += red[wv][threadIdx.x];
        part[(size_t)wg * 64 + threadIdx.x] = s;   // [wg][k][c]
    }
}

// ---------------------------------------------------------------------------
// K2: one wave per (b,k): finalize xy moments, z softmax gather, pose output
// ---------------------------------------------------------------------------
__global__ __launch_bounds__(128) void k2_per_kp(const float* __restrict__ feat,
                                                 const float* __restrict__ gt,
                                                 const float* __restrict__ part,
                                                 float* __restrict__ loss4,   // [480][4]
                                                 float* __restrict__ pose)    // [480][3]
{
    const int lane = threadIdx.x & 31;
    const int item = blockIdx.x * 4 + (threadIdx.x >> 5);  // 0..479 exactly
    const int b = item / KPT;
    const int k = item % KPT;

    // ---- combine NCH partial sums: lane = ch*4 + c ----
    const int c    = lane & 3;
    const int chnk = lane >> 2;
    float v = part[((size_t)(b * NCH + chnk)) * 64 + k * 4 + c];
    v += __shfl_xor(v, 4);
    v += __shfl_xor(v, 8);
    v += __shfl_xor(v, 16);
    const float S0 = __shfl(v, 0);
    const float Sx = __shfl(v, 1);
    const float Sy = __shfl(v, 2);
    const float Sr = __shfl(v, 3);
    const float inv = 1.0f / S0;
    const float lx = Sx * inv, ly = Sy * inv;
    const float var_xy = Sr * inv - (lx * lx + ly * ly);

    // ---- gt + pixel indices: lanes 0..14 handle k' = lane ----
    float gx = 0.f, gy = 0.f, gz = 0.f;
    int hh = -1, ww = -1;
    if (lane < KPT) {
        gx = gt[(b * KPT + lane) * 3 + 0];
        gy = gt[(b * KPT + lane) * 3 + 1];
        gz = gt[(b * KPT + lane) * 3 + 2];
        const float cgx = fminf(fmaxf(gx, MINL), 150.0f);
        const float cgy = fminf(fmaxf(gy, MINL), 150.0f);
        hh = (int)fminf(fmaxf(rintf((cgx - MINL) / DXY), 0.0f), 63.0f);
        ww = (int)fminf(fmaxf(rintf((cgy - MINL) / DXY), 0.0f), 63.0f);
    }
    const int h_me = __shfl(hh, k);
    const int w_me = __shfl(ww, k);

    // scatter-with-duplicates semantics: last k' writing this pixel wins
    int cand = (lane < KPT && hh == h_me && ww == w_me) ? lane : -1;
    cand = max(cand, __shfl_xor(cand, 1));
    cand = max(cand, __shfl_xor(cand, 2));
    cand = max(cand, __shfl_xor(cand, 4));
    cand = max(cand, __shfl_xor(cand, 8));
    cand = max(cand, __shfl_xor(cand, 16));
    const float gt_z = fminf(fmaxf(__shfl(gz, cand), MINL), 150.0f);

    const float gtx = fminf(fmaxf(__shfl(gx, k), MINL), 150.0f);
    const float gty = fminf(fmaxf(__shfl(gy, k), MINL), 150.0f);
    const float pos_xy = (lx - gtx) * (lx - gtx) + (ly - gty) * (ly - gty);

    // ---- z softmax at the gathered pixel: 2 bins per lane ----
    const float* fz = feat + (((size_t)(b * HH + h_me) * WW + w_me) * CHN + KPT);
    const float z0 = MINL + lane * DXY;
    const float z1 = MINL + (lane + 32) * DXY;
    const float e0 = __expf(fz[lane]);
    const float e1 = __expf(fz[lane + 32]);
    float s   = e0 + e1;
    float sz  = e0 * z0 + e1 * z1;
    float szz = e0 * z0 * z0 + e1 * z1 * z1;
    #pragma unroll
    for (int msk = 1; msk < 32; msk <<= 1) {
        s   += __shfl_xor(s, msk);
        sz  += __shfl_xor(sz, msk);
        szz += __shfl_xor(szz, msk);
    }
    const float invz  = 1.0f / s;
    const float lz    = sz * invz;
    const float var_z = szz * invz - lz * lz;
    const float pos_z = (lz - gt_z) * (lz - gt_z);

    if (lane == 0) {
        loss4[item * 4 + 0] = var_xy;
        loss4[item * 4 + 1] = pos_xy;
        loss4[item * 4 + 2] = var_z;
        loss4[item * 4 + 3] = pos_z;
        pose[item * 3 + 0] = lx;
        pose[item * 3 + 1] = ly;
        pose[item * 3 + 2] = lz;
    }
}

// ---------------------------------------------------------------------------
// K3: single workgroup finalize: hard-kp factors + limb/sym + total loss
//   sum(loss*factor) = sum_{k,c} colsum(k,c)^2 * K / (s * 150)
// ---------------------------------------------------------------------------
__global__ __launch_bounds__(512) void k3_final(const float* __restrict__ gt,
                                                const float* __restrict__ loss4,
                                                const float* __restrict__ pose,
                                                float* __restrict__ out_total)
{
    __shared__ float lp[BB * KPT];
    __shared__ float cs[64];
    __shared__ float red[512];
    const int t = threadIdx.x;

    float myterm = 0.0f;
    if (t < BB * KPT) {
        const int b = t / KPT, k = t % KPT;
        const int pk = d_limb[k];
        const int pj = b * KPT + pk;
        float dx = pose[t * 3 + 0] - pose[pj * 3 + 0];
        float dy = pose[t * 3 + 1] - pose[pj * 3 + 1];
        float dz = pose[t * 3 + 2] - pose[pj * 3 + 2];
        const float l = sqrtf(dx * dx + dy * dy + dz * dz + 1e-12f);
        lp[t] = l;
        dx = gt[t * 3 + 0] - gt[pj * 3 + 0];
        dy = gt[t * 3 + 1] - gt[pj * 3 + 1];
        dz = gt[t * 3 + 2] - gt[pj * 3 + 2];
        const float lg = sqrtf(dx * dx + dy * dy + dz * dz + 1e-12f);
        myterm = (l - lg) * (l - lg);
    }
    if (t < 60) {   // (k,c) column sums over b
        const int k = t >> 2, c = t & 3;
        float acc = 0.0f;
        for (int b = 0; b < BB; ++b) acc += loss4[(b * KPT + k) * 4 + c];
        cs[t] = acc;
    }
    __syncthreads();
    if (t < BB * KPT) {
        const int b = t / KPT, k = t % KPT;
        const float d = lp[t] - lp[b * KPT + d_sym[k]];
        myterm += d * d;
    }
    red[t] = myterm;
    __syncthreads();
    #pragma unroll
    for (int stride = 256; stride > 0; stride >>= 1) {
        if (t < stride) red[t] += red[t + stride];
        __syncthreads();
    }
    if (t == 0) {
        float s_xy = 0.001f, s_z = 0.001f, q_xy = 0.0f, q_z = 0.0f;
        for (int i = 0; i < 60; ++i) {
            const int c = i & 3;
            const float v = cs[i];
            if (c < 2) { s_xy += v; q_xy += v * v; }
            else       { s_z  += v; q_z  += v * v; }
        }
        const float total = red[0]
            + q_xy * (float)KPT / (s_xy * 150.0f)
            + q_z  * (float)KPT / (s_z  * 150.0f);
        out_total[0] = total;
    }
}

// ---------------------------------------------------------------------------
extern "C" void kernel_launch(void* const* d_in, const int* in_sizes, int n_in,
                              void* d_out, int out_size, void* d_ws, size_t ws_size,
                              hipStream_t stream) {
    const float* feat = (const float*)d_in[0];   // (32,64,64,79) f32
    const float* gt   = (const float*)d_in[1];   // (32,15,3) f32
    float* out   = (float*)d_out;                // [total, pose(1440)]
    float* part  = (float*)d_ws;                 // [BB*NCH][64] = 64 KB
    float* loss4 = part + (size_t)BB * NCH * 64; // [480][4]

    k1_xy_partial<<<BB * NCH, 256, 0, stream>>>(feat, part);
    k2_per_kp<<<(BB * KPT) / 4, 128, 0, stream>>>(feat, gt, part, loss4, out + 1);
    k3_final<<<1, 512, 0, stream>>>(gt, loss4, out + 1, out);
}